// EdgeDecoder_22789096472976
// MI455X (gfx1250) — compile-verified
//
#include <hip/hip_runtime.h>
#include <hip/hip_bf16.h>
#include <stdint.h>

#define HIDDEN 128
#define TWO_H  256
#define PAD2   144   // float2 row pitch in LDS: 144*2 dwords % 64 banks = 32 -> halves hit disjoint banks

typedef __attribute__((ext_vector_type(2))) float v2f;
typedef __attribute__((ext_vector_type(8))) float v8f;

// One wave (32 lanes) computes a 16-edge tile:
//   A (16x256, f32, gathered rows of z) x B (256x128 = W1^T) -> C (16x128) via V_WMMA_F32_16X16X4_F32,
// then fused bias1 + ReLU + dot(W2) + bias2 reduction -> 16 scalars.
__global__ __launch_bounds__(256)
void edge_decoder_wmma_f32(const float* __restrict__ z,
                           const float* __restrict__ W1,
                           const float* __restrict__ bias1,
                           const float* __restrict__ W2,
                           const float* __restrict__ bias2,
                           const int*   __restrict__ edge_index,  // [2, E]
                           float*       __restrict__ out,
                           int n_edges)
{
    extern __shared__ v2f lds[];   // [128 pairs][PAD2 cols]: lds[p*PAD2 + n] = {W1T[2p][n], W1T[2p+1][n]}

    const int tid = threadIdx.x;

    // ---- Stage W1^T into LDS as k-pairs (coalesced 8B reads along W1 rows) ----
    for (int idx = tid; idx < HIDDEN * (TWO_H / 2); idx += blockDim.x) {
        const int p = idx & 127;          // k-pair index 0..127
        const int n = idx >> 7;           // output column 0..127
        v2f v;
        v.x = W1[n * TWO_H + 2 * p];      // W1 is [128][256] row-major; B[k][n] = W1[n][k]
        v.y = W1[n * TWO_H + 2 * p + 1];
        lds[p * PAD2 + n] = v;
    }
    __syncthreads();

    const int lane = tid & 31;
    const int wave = tid >> 5;
    const int m    = lane & 15;           // row within 16-edge tile / column within 16-wide N-tile
    const int half = lane >> 4;           // 0: lanes 0-15, 1: lanes 16-31

    // Per-lane epilogue constants (column n = nt*16 + m)
    float b1v[8], w2v[8];
#pragma unroll
    for (int nt = 0; nt < 8; ++nt) {
        b1v[nt] = bias1[nt * 16 + m];
        w2v[nt] = W2[nt * 16 + m];
    }
    const float b2v = bias2[0];

    const int n_tiles = n_edges >> 4;     // E = 640000 -> 40000 tiles of 16 edges

    for (int tile = blockIdx.x * 8 + wave; tile < n_tiles; tile += gridDim.x * 8) {
        const int e0 = tile * 16;
        // Per-lane gather base pointers for row M = m of the A tile
        const int si = edge_index[e0 + m];                 // edge_index[0][e]
        const int di = edge_index[n_edges + e0 + m];       // edge_index[1][e]
        const float* __restrict__ psrc = z + (long)si * HIDDEN;
        const float* __restrict__ pdst = z + (long)di * HIDDEN;

        v8f acc[8];
#pragma unroll
        for (int nt = 0; nt < 8; ++nt) acc[nt] = v8f{};

#pragma unroll 2
        for (int k0 = 0; k0 < TWO_H; k0 += 4) {
            // A fragment (16x4 f32): lane (half*16+m) holds (M=m, K=kk) and (M=m, K=kk+1)
            const int kk = k0 + half * 2;
            const float* __restrict__ pa = (kk < HIDDEN) ? (psrc + kk) : (pdst + (kk - HIDDEN));
            v2f a;
            a.x = pa[0];
            a.y = pa[1];

            const int p = kk >> 1;        // k-pair index in LDS (kk is even; pair never straddles 128)
#pragma unroll
            for (int nt = 0; nt < 8; ++nt) {
                // B fragment (4x16 f32): lane holds (K=kk, N=nt*16+m) and (K=kk+1, N=...)
                v2f b = lds[p * PAD2 + nt * 16 + m];
                acc[nt] = __builtin_amdgcn_wmma_f32_16x16x4_f32(
                    /*neg_a=*/false, a, /*neg_b=*/false, b,
                    /*c_mod=*/(short)0, acc[nt],
                    /*reuse_a=*/false, /*reuse_b=*/false);
            }
        }

        // ---- Fused epilogue: relu(C + bias1) . W2 + bias2, reduced over N ----
        // C layout: VGPR j of lane: row M = j + half*8, column N = (lane&15) ... per 16-wide tile
        float red[8];
#pragma unroll
        for (int j = 0; j < 8; ++j) {
            float s = 0.f;
#pragma unroll
            for (int nt = 0; nt < 8; ++nt) {
                float h = acc[nt][j] + b1v[nt];
                h = fmaxf(h, 0.f);
                s = fmaf(h, w2v[nt], s);
            }
            // reduce the 16 columns held by each 16-lane half (xor masks stay inside the half)
            s += __shfl_xor(s, 1, 32);
            s += __shfl_xor(s, 2, 32);
            s += __shfl_xor(s, 4, 32);
            s += __shfl_xor(s, 8, 32);
            red[j] = s;
        }
        if (m == 0) {                     // lane 0 -> rows 0..7, lane 16 -> rows 8..15
            const int base = e0 + half * 8;
#pragma unroll
            for (int j = 0; j < 8; ++j) out[base + j] = red[j] + b2v;
        }
    }
}

extern "C" void kernel_launch(void* const* d_in, const int* in_sizes, int n_in,
                              void* d_out, int out_size, void* d_ws, size_t ws_size,
                              hipStream_t stream) {
    const float* z     = (const float*)d_in[0];   // [50000,128]
    const float* W1    = (const float*)d_in[1];   // [128,256]
    const float* bias1 = (const float*)d_in[2];   // [128]
    const float* W2    = (const float*)d_in[3];   // [1,128]
    const float* bias2 = (const float*)d_in[4];   // [1]
    const int*   ei    = (const int*)d_in[5];     // [2,E]
    float* out = (float*)d_out;

    const int n_edges = out_size;                 // 640000
    const int n_tiles = n_edges >> 4;             // 16 edges per wave-tile

    const size_t ldsBytes = (size_t)128 * PAD2 * sizeof(v2f);   // 147456 B <= 320 KB/WGP

    // Allow >64KB dynamic LDS (idempotent, deterministic; not a stream op)
    hipFuncSetAttribute((const void*)edge_decoder_wmma_f32,
                        hipFuncAttributeMaxDynamicSharedMemorySize, (int)ldsBytes);

    // 8 wave-tiles per block; ~4 tiles per wave to amortize the LDS staging of W1^T
    int blocks = (n_tiles + 8 * 4 - 1) / (8 * 4);
    if (blocks < 1) blocks = 1;

    edge_decoder_wmma_f32<<<blocks, 256, ldsBytes, stream>>>(
        z, W1, bias1, W2, bias2, ei, out, n_edges);
}